// FancyMLP_13082470384150
// MI455X (gfx1250) — compile-verified
//
#include <hip/hip_runtime.h>
#include <hip/hip_bf16.h>

typedef float v2f __attribute__((ext_vector_type(2)));
typedef float v8f __attribute__((ext_vector_type(8)));

#define NBLK 1024          // blocks for main kernel
#define NTILES 62500       // 1,000,000 rows / 16 rows per tile (exact)
#define D 20

// ---------------------------------------------------------------------------
// Main kernel: each wave (32 lanes) processes 16-row tiles of X.
//   GEMM1: X1 = X @ W^T + b          (K = 20 = 5 chunks of 4)
//   GEMM2: X2 = relu(X1 @ R + 1)
//   GEMM3: X3 = X2 @ W^T + b
// Per-thread partial sums of X3 and |X3| -> per-block partials in d_ws.
// ---------------------------------------------------------------------------
__global__ __launch_bounds__(256) void fancy_mlp_main(
    const float* __restrict__ X, const float* __restrict__ W,
    const float* __restrict__ bvec, const float* __restrict__ R,
    float* __restrict__ partials) {
  const int tid  = threadIdx.x;
  const int l    = tid & 31;        // lane in wave
  const int w    = tid >> 5;        // wave in block (8 waves)
  const int m    = l & 15;          // lane row/col index
  const int half = l >> 4;          // 0: lanes 0-15, 1: lanes 16-31

  // Per-wave LDS slab: stage X tile (16x20 packed) then X1/X2 (16x32).
  __shared__ __align__(16) float lds[8][512];
  __shared__ float redS[256], redA[256];
  float* st = lds[w];

  // ---- Wave-uniform B fragments in registers (loaded once) ----
  // B layout for 16x16x4 f32: VGPR0/1 hold rows K=kc+2*half, +1; N = lane&15.
  v2f bW0[5], bW1[5], bR0[5], bR1[5];
  const int n0 = m;            // n-tile 0: cols 0..15, always valid
  const int n1 = 16 + m;       // n-tile 1: cols 16..31, valid iff < 20
  const bool v1ok = (n1 < D);
#pragma unroll
  for (int i = 0; i < 5; ++i) {
    const int k = i * 4 + 2 * half;              // k, k+1 both < 20
    // GEMM1/3: B = W^T, so B[k][n] = W[n][k]
    bW0[i].x = W[n0 * D + k];
    bW0[i].y = W[n0 * D + k + 1];
    bW1[i].x = v1ok ? W[n1 * D + k]     : 0.0f;
    bW1[i].y = v1ok ? W[n1 * D + k + 1] : 0.0f;
    // GEMM2: B = R, so B[k][n] = R[k][n]
    bR0[i].x = R[k * D + n0];
    bR0[i].y = R[(k + 1) * D + n0];
    bR1[i].x = v1ok ? R[k * D + n1]       : 0.0f;
    bR1[i].y = v1ok ? R[(k + 1) * D + n1] : 0.0f;
  }
  const float bias0 = bvec[m];
  const float bias1 = v1ok ? bvec[n1] : 0.0f;

  float sum = 0.0f, asum = 0.0f;

  for (int tile = blockIdx.x * 8 + w; tile < NTILES; tile += gridDim.x * 8) {
    // ---- Stage 16x20 X tile: 1280 contiguous bytes, fully coalesced ----
    const float* src = X + (size_t)tile * (16 * D);
    float4 p0 = *(const float4*)(src + 4 * l);
    float4 p1 = *(const float4*)(src + 128 + 4 * l);
    float2 p2 = *(const float2*)(src + 256 + 2 * l);
    *(float4*)(st + 4 * l)       = p0;
    *(float4*)(st + 128 + 4 * l) = p1;
    *(float2*)(st + 256 + 2 * l) = p2;

    // ---- A fragments of X (row m, packed stride 20) ----
    v2f a[5];
#pragma unroll
    for (int i = 0; i < 5; ++i)
      a[i] = *(const v2f*)(st + m * D + i * 4 + 2 * half);

    // ---- GEMM1: C = X @ W^T ----
    v8f c0 = {0,0,0,0,0,0,0,0}, c1 = {0,0,0,0,0,0,0,0};
#pragma unroll
    for (int i = 0; i < 5; ++i) {
      c0 = __builtin_amdgcn_wmma_f32_16x16x4_f32(false, a[i], false, bW0[i],
                                                 (short)0, c0, false, false);
      c1 = __builtin_amdgcn_wmma_f32_16x16x4_f32(false, a[i], false, bW1[i],
                                                 (short)0, c1, false, false);
    }
    // X1 = C + b  -> LDS (16x32, padded cols are exactly zero)
#pragma unroll
    for (int r = 0; r < 8; ++r) {
      const int row = r + 8 * half;
      st[row * 32 + m]      = c0[r] + bias0;
      st[row * 32 + 16 + m] = c1[r] + bias1;
    }

    // ---- A fragments of X1 (stride 32) ----
#pragma unroll
    for (int i = 0; i < 5; ++i)
      a[i] = *(const v2f*)(st + m * 32 + i * 4 + 2 * half);

    // ---- GEMM2: D = X1 @ R ----
    v8f d0 = {0,0,0,0,0,0,0,0}, d1 = {0,0,0,0,0,0,0,0};
#pragma unroll
    for (int i = 0; i < 5; ++i) {
      d0 = __builtin_amdgcn_wmma_f32_16x16x4_f32(false, a[i], false, bR0[i],
                                                 (short)0, d0, false, false);
      d1 = __builtin_amdgcn_wmma_f32_16x16x4_f32(false, a[i], false, bR1[i],
                                                 (short)0, d1, false, false);
    }
    // X2 = relu(D + 1), masked so padded cols stay zero -> LDS
#pragma unroll
    for (int r = 0; r < 8; ++r) {
      const int row = r + 8 * half;
      st[row * 32 + m]      = fmaxf(d0[r] + 1.0f, 0.0f);
      st[row * 32 + 16 + m] = v1ok ? fmaxf(d1[r] + 1.0f, 0.0f) : 0.0f;
    }

    // ---- A fragments of X2 ----
#pragma unroll
    for (int i = 0; i < 5; ++i)
      a[i] = *(const v2f*)(st + m * 32 + i * 4 + 2 * half);

    // ---- GEMM3: E = X2 @ W^T ----
    v8f e0 = {0,0,0,0,0,0,0,0}, e1 = {0,0,0,0,0,0,0,0};
#pragma unroll
    for (int i = 0; i < 5; ++i) {
      e0 = __builtin_amdgcn_wmma_f32_16x16x4_f32(false, a[i], false, bW0[i],
                                                 (short)0, e0, false, false);
      e1 = __builtin_amdgcn_wmma_f32_16x16x4_f32(false, a[i], false, bW1[i],
                                                 (short)0, e1, false, false);
    }
    // X3 = E + b; padded cols give exactly 0 (zero W^T cols, zero bias),
    // so they contribute nothing to either reduction.
#pragma unroll
    for (int r = 0; r < 8; ++r) {
      const float xa = e0[r] + bias0;
      const float xb = e1[r] + bias1;
      sum  += xa + xb;
      asum += fabsf(xa) + fabsf(xb);
    }
  }

  // ---- Deterministic in-block tree reduction ----
  redS[tid] = sum;
  redA[tid] = asum;
  __syncthreads();
#pragma unroll
  for (int s = 128; s > 0; s >>= 1) {
    if (tid < s) {
      redS[tid] += redS[tid + s];
      redA[tid] += redA[tid + s];
    }
    __syncthreads();
  }
  if (tid == 0) {
    partials[blockIdx.x]        = redS[0];
    partials[NBLK + blockIdx.x] = redA[0];
  }
}

// ---------------------------------------------------------------------------
// Final kernel: deterministic reduction of per-block partials, then the
// algebraic halving loop (each halve scales S and S_abs by exactly 0.5).
// ---------------------------------------------------------------------------
__global__ __launch_bounds__(256) void fancy_mlp_finish(
    const float* __restrict__ partials, float* __restrict__ out) {
  __shared__ float sS[256], sA[256];
  const int tid = threadIdx.x;
  float s = 0.0f, a = 0.0f;
  for (int j = tid; j < NBLK; j += 256) {   // fixed traversal order
    s += partials[j];
    a += partials[NBLK + j];
  }
  sS[tid] = s;
  sA[tid] = a;
  __syncthreads();
#pragma unroll
  for (int st = 128; st > 0; st >>= 1) {
    if (tid < st) {
      sS[tid] += sS[tid + st];
      sA[tid] += sA[tid + st];
    }
    __syncthreads();
  }
  if (tid == 0) {
    float S = sS[0], A = sA[0];
    // while sum(|X|) > 1: X /= 2   (300 iters covers any finite fp32)
    for (int it = 0; it < 300 && A > 1.0f; ++it) {
      A *= 0.5f;
      S *= 0.5f;
    }
    out[0] = S;
  }
}

extern "C" void kernel_launch(void* const* d_in, const int* in_sizes, int n_in,
                              void* d_out, int out_size, void* d_ws, size_t ws_size,
                              hipStream_t stream) {
  const float* X = (const float*)d_in[0];  // (1e6, 20)
  const float* W = (const float*)d_in[1];  // (20, 20)
  const float* b = (const float*)d_in[2];  // (20,)
  const float* R = (const float*)d_in[3];  // (20, 20)
  float* partials = (float*)d_ws;          // 2 * NBLK floats = 8 KB
  (void)in_sizes; (void)n_in; (void)out_size; (void)ws_size;

  fancy_mlp_main<<<NBLK, 256, 0, stream>>>(X, W, b, R, partials);
  fancy_mlp_finish<<<1, 256, 0, stream>>>(partials, (float*)d_out);
}